// CFConv_48687749267992
// MI455X (gfx1250) — compile-verified
//
#include <hip/hip_runtime.h>
#include <stdint.h>

// CFConv: y[idx_i[e]] += x[idx_j[e]] * Wij[e]   (E=3.2M, F=128, fp32, idx_i sorted)
//
// Memory-bound (0.47 FLOP/B) -> optimized for the 23.3 TB/s Wij stream:
//  - Wij staged into LDS via CDNA5 GLOBAL_LOAD_ASYNC_TO_LDS_B128 (ASYNCcnt,
//    double-buffered, th:TH_LOAD_NT so the 1.64 GB one-shot stream doesn't
//    evict the L2-resident x (51 MB) and y (51 MB) working sets).
//  - wave32 lane l owns features [4l,4l+4): one wave = one full F=128 row,
//    all traffic is b128.
//  - sorted idx_i -> run-length accumulate in registers, flush with
//    global_atomic_add_f32 only at segment boundaries.

typedef float vfloat4 __attribute__((ext_vector_type(4)));

#define F_DIM       128
#define WAVES       8                   // 256 threads / block (8 wave32)
#define EPS         8                   // edges per pipeline stage per wave
#define STAGES      8                   // stages per wave -> 64 edges per wave
#define ROW_BYTES   (F_DIM * 4)         // 512 B per edge row (32 lanes x 16B)
#define STAGE_BYTES (EPS * ROW_BYTES)   // 4 KB
#define WAVE_LDS    (2 * STAGE_BYTES)   // 8 KB (double buffer)
#define BLOCK_LDS   (WAVES * WAVE_LDS)  // 64 KB static LDS
#define EDGES_PER_WAVE  (EPS * STAGES)              // 64
#define EDGES_PER_BLOCK (WAVES * EDGES_PER_WAVE)    // 512

__device__ __forceinline__ vfloat4 vzero4() {
  vfloat4 z; z.x = 0.f; z.y = 0.f; z.z = 0.f; z.w = 0.f; return z;
}

__device__ __forceinline__ void atomic_flush(float* __restrict__ y, int seg, int lane, vfloat4 a) {
  float* p = y + (size_t)seg * F_DIM + (size_t)lane * 4;
  __hip_atomic_fetch_add(p + 0, a.x, __ATOMIC_RELAXED, __HIP_MEMORY_SCOPE_AGENT);
  __hip_atomic_fetch_add(p + 1, a.y, __ATOMIC_RELAXED, __HIP_MEMORY_SCOPE_AGENT);
  __hip_atomic_fetch_add(p + 2, a.z, __ATOMIC_RELAXED, __HIP_MEMORY_SCOPE_AGENT);
  __hip_atomic_fetch_add(p + 3, a.w, __ATOMIC_RELAXED, __HIP_MEMORY_SCOPE_AGENT);
}

// Zero-init d_out (harness poisons it with 0xAA before timing).
__global__ void __launch_bounds__(256) cfconv_zero(float* __restrict__ out, long n) {
  const long n4 = n >> 2;
  const long i0 = (long)blockIdx.x * blockDim.x + threadIdx.x;
  const long stride = (long)gridDim.x * blockDim.x;
  const vfloat4 z = vzero4();
  for (long k = i0; k < n4; k += stride) ((vfloat4*)out)[k] = z;
  const long tail = n4 << 2;
  if (blockIdx.x == 0 && (tail + threadIdx.x) < n) out[tail + threadIdx.x] = 0.f;
}

// Fast path: wave's 64-edge window entirely in range -> no per-edge checks.
__device__ __forceinline__ void cfconv_wave_full(
    const float* __restrict__ x, const float* __restrict__ Wij,
    const int* __restrict__ idx_i, const int* __restrict__ idx_j,
    float* __restrict__ y, long base, int lane,
    char* mybuf, uint32_t lds_base)
{
  // Issue one pipeline stage: EPS async b128 row copies (32 lanes x 16B each).
  // NT temporal hint: Wij is a one-shot stream; keep it out of L2.
  // No "memory" clobber here: volatile asm keeps program order among asms,
  // and the LDS data is only consumed after the s_wait_asynccnt asm (which
  // does carry the clobber). This keeps the compiler free to schedule the
  // scalar idx loads / x gathers around the issue burst.
  auto issue = [&](int sbuf, long ebase) {
    const uint32_t l0 = lds_base + (uint32_t)sbuf * STAGE_BYTES + (uint32_t)lane * 16u;
#pragma unroll
    for (int t = 0; t < EPS; ++t) {
      const float* wrow = Wij + (ebase + t) * F_DIM;   // uniform base; 64-bit VADDR form, saddr off
      const uint32_t la = l0 + (uint32_t)t * ROW_BYTES;
      asm volatile("global_load_async_to_lds_b128 %0, %1, off th:TH_LOAD_NT"
                   :: "v"(la), "v"(wrow + lane * 4));
    }
  };

  issue(0, base);                       // prime the pipeline

  vfloat4 acc = vzero4();
  int cur = idx_i[base];                // wave-uniform -> scalar (SMEM) load

#pragma unroll
  for (int s = 0; s < STAGES; ++s) {
    const long sbase = base + (long)s * EPS;
    if (s + 1 < STAGES) {
      issue((s + 1) & 1, sbase + EPS);                           // prefetch next stage
      asm volatile("s_wait_asynccnt %0" :: "n"(EPS) : "memory"); // stage s landed
    } else {
      asm volatile("s_wait_asynccnt 0" ::: "memory");
    }

    const char* stg = mybuf + (s & 1) * STAGE_BYTES;
#pragma unroll
    for (int t = 0; t < EPS; ++t) {
      const long e = sbase + t;
      const int seg = idx_i[e];                     // uniform -> s_load
      const int j   = idx_j[e];                     // uniform -> s_load
      const float* xrow = x + (size_t)j * F_DIM;    // uniform base -> SADDR form
      const vfloat4 w  = *(const vfloat4*)(stg + t * ROW_BYTES + lane * 16); // ds_load_b128
      const vfloat4 xv = *(const vfloat4*)(xrow + lane * 4);                 // L2 gather
      if (seg != cur) {                             // segment boundary: flush run
        atomic_flush(y, cur, lane, acc);
        acc = vzero4();
        cur = seg;
      }
      acc += xv * w;                                // packed fp32 FMA
    }
  }
  atomic_flush(y, cur, lane, acc);                  // final run
}

// Ragged last wave: simple direct-load path with per-edge bounds.
__device__ __forceinline__ void cfconv_wave_tail(
    const float* __restrict__ x, const float* __restrict__ Wij,
    const int* __restrict__ idx_i, const int* __restrict__ idx_j,
    float* __restrict__ y, int E, long base, int lane)
{
  vfloat4 acc = vzero4();
  int cur = idx_i[base];
  const long nloc = (long)E - base;                 // 1..EDGES_PER_WAVE-1 edges
  for (long t = 0; t < nloc; ++t) {
    const long e = base + t;
    const int seg = idx_i[e];
    const int j   = idx_j[e];
    const float* wrow = Wij + e * F_DIM;
    const float* xrow = x + (size_t)j * F_DIM;
    const vfloat4 w  = *(const vfloat4*)(wrow + lane * 4);
    const vfloat4 xv = *(const vfloat4*)(xrow + lane * 4);
    if (seg != cur) {
      atomic_flush(y, cur, lane, acc);
      acc = vzero4();
      cur = seg;
    }
    acc += xv * w;
  }
  atomic_flush(y, cur, lane, acc);
}

__global__ void __launch_bounds__(256) cfconv_kernel(
    const float* __restrict__ x, const float* __restrict__ Wij,
    const int* __restrict__ idx_i, const int* __restrict__ idx_j,
    float* __restrict__ y, int E)
{
  __shared__ __align__(16) char smem[BLOCK_LDS];

  const int lane = threadIdx.x & 31;
  const int wave = threadIdx.x >> 5;
  const long wid  = (long)blockIdx.x * WAVES + wave;
  const long base = wid * (long)EDGES_PER_WAVE;
  if (base >= (long)E) return;

  char* mybuf = smem + wave * WAVE_LDS;
  // addr[31:0] of a flat pointer into the LDS aperture is the wave-relative
  // LDS byte offset (CDNA5 aperture mapping) -> usable as async VDST address.
  const uint32_t lds_base = (uint32_t)(uintptr_t)mybuf;

  if (base + EDGES_PER_WAVE <= (long)E) {
    cfconv_wave_full(x, Wij, idx_i, idx_j, y, base, lane, mybuf, lds_base);
  } else {
    cfconv_wave_tail(x, Wij, idx_i, idx_j, y, E, base, lane);
  }
}

extern "C" void kernel_launch(void* const* d_in, const int* in_sizes, int n_in,
                              void* d_out, int out_size, void* d_ws, size_t ws_size,
                              hipStream_t stream) {
  const float* x     = (const float*)d_in[0];
  const float* Wij   = (const float*)d_in[1];
  const int*   idx_i = (const int*)d_in[2];
  const int*   idx_j = (const int*)d_in[3];
  float*       y     = (float*)d_out;
  const int E = in_sizes[2];
  const long n = (long)out_size;

  const long n4 = n >> 2;
  int zblocks = (int)((n4 + 255) / 256);
  if (zblocks < 1) zblocks = 1;
  cfconv_zero<<<zblocks, 256, 0, stream>>>(y, n);

  const int mblocks = (int)(((long)E + EDGES_PER_BLOCK - 1) / EDGES_PER_BLOCK);
  cfconv_kernel<<<mblocks, 256, 0, stream>>>(x, Wij, idx_i, idx_j, y, E);
}